// Net_80736795230776
// MI455X (gfx1250) — compile-verified
//
#include <hip/hip_runtime.h>
#include <cmath>

// Problem constants (match the reference).
#define IN_DIM  135909
#define OUT_DIM 670091
#define EMB_D   128
#define BATCH   256
#define TFEAT   100

#define NCHUNK  256                                   // classes per block in the GEMM pass
#define NBLK    ((OUT_DIM + NCHUNK - 1) / NCHUNK)     // 2618 blocks
#define NTILES  (NCHUNK / 16)                         // 16 N-tiles per block

#define TSTRIDE 132                                   // padded LDS row stride (floats)
#define TILE_F  (16 * TSTRIDE)                        // one staged W tile in LDS (floats)

typedef __attribute__((ext_vector_type(2))) float v2f;
typedef __attribute__((ext_vector_type(4))) float v4f;
typedef __attribute__((ext_vector_type(8))) float v8f;

// ---------------------------------------------------------------------------
// Kernel 1: bag-of-features embedding sum, L2 normalize, +bias, ReLU -> query
// grid = BATCH blocks, block = 128 threads (one per dim)
// ---------------------------------------------------------------------------
__global__ __launch_bounds__(EMB_D)
void embed_query_kernel(const int* __restrict__ x,
                        const float* __restrict__ emb,
                        const float* __restrict__ bias,
                        float* __restrict__ q)
{
    const int b = blockIdx.x;
    const int d = threadIdx.x;
    const int* xb = x + b * TFEAT;

    float acc = 0.f;
    #pragma unroll 4
    for (int t = 0; t < TFEAT; ++t)
        acc += emb[(long)xb[t] * EMB_D + d];

    __shared__ float red[EMB_D];
    red[d] = acc * acc;
    __syncthreads();
    for (int off = EMB_D / 2; off > 0; off >>= 1) {
        if (d < off) red[d] += red[d + off];
        __syncthreads();
    }
    const float rn = 1.0f / sqrtf(red[0]);
    q[b * EMB_D + d] = fmaxf(acc * rn + bias[d], 0.0f);
}

// ---------------------------------------------------------------------------
// Kernel 2: fused logits GEMM (query @ W^T + b_out) + online softmax partials.
// grid = NBLK blocks (256 classes each), block = 256 threads = 8 wave32.
// Wave w computes batch rows [32w, 32w+32) as two 16x16x(K=128) WMMA tiles.
// W tiles are staged through double-buffered LDS (each W line is touched by
// exactly one block grid-wide -> NT global loads; LDS serves the 8 waves).
// Emits per-(block,row) partial (max, sumexp) for log-sum-exp combining.
// ---------------------------------------------------------------------------
__global__ __launch_bounds__(256)
void gemm_softmax_stats_kernel(const float* __restrict__ q,
                               const float* __restrict__ W,
                               const float* __restrict__ bout,
                               float* __restrict__ pm,
                               float* __restrict__ ps)
{
    __shared__ float ldsb[2 * TILE_F];  // double-buffered 16x128 W tile (padded rows)
    __shared__ float smx[BATCH * 16];   // per-row, per-half-lane partial max
    __shared__ float sms[BATCH * 16];   // per-row, per-half-lane partial sumexp

    const int tid  = threadIdx.x;
    const int lane = tid & 31;
    const int wave = tid >> 5;            // 0..7
    const int nsub = lane & 15;           // M (A) / N (B,C,D) index of this lane
    const int kb   = (lane >> 4) * 2;     // K pair selected by lane[4] (ISA f32 layout)
    const int rowbase = wave * 32;

    // Preload A fragments for both M tiles across the full K=128 (32 steps of 4).
    // Lane holds A[m][k0+kb], A[m][k0+kb+1] per step -> contiguous float2.
    v2f a0[32], a1[32];
    {
        const float* q0 = q + (rowbase + nsub) * EMB_D + kb;
        const float* q1 = q0 + 16 * EMB_D;
        #pragma unroll
        for (int j = 0; j < 32; ++j) {
            a0[j] = *(const v2f*)(q0 + 4 * j);
            a1[j] = *(const v2f*)(q1 + 4 * j);
        }
    }

    // Per-lane online softmax state: 8 row-slots per M tile (C/D layout:
    // VGPR r -> row r (+8 for lanes 16..31), lane&15 -> column).
    float m0[8], s0[8], m1[8], s1[8];
    #pragma unroll
    for (int r = 0; r < 8; ++r) {
        m0[r] = -INFINITY; s0[r] = 0.f;
        m1[r] = -INFINITY; s1[r] = 0.f;
    }

    const int n0blk = blockIdx.x * NCHUNK;

    // Cooperative tile staging: thread t handles 8 consecutive floats of the
    // 16x128 tile: tile row = t/16, tile cols [ (t%16)*8, +8 ).
    const int trow = tid >> 4;
    const int tcol = (tid & 15) * 8;
    const int dsoff = trow * TSTRIDE + tcol;

    // Prologue: fetch tile 0 into registers.
    v4f r0, r1;
    {
        const long grow = min((long)(n0blk + trow), (long)(OUT_DIM - 1));
        const float* src = W + grow * EMB_D + tcol;
        r0 = __builtin_nontemporal_load((const v4f*)src);
        r1 = __builtin_nontemporal_load((const v4f*)(src + 4));
    }

    for (int nt = 0; nt < NTILES; ++nt) {
        const int buf = (nt & 1) * TILE_F;

        // Commit the prefetched tile to LDS.
        *(v4f*)&ldsb[buf + dsoff]     = r0;
        *(v4f*)&ldsb[buf + dsoff + 4] = r1;

        // Issue next tile's global loads (waited on at next iteration's store).
        if (nt + 1 < NTILES) {
            const long grow = min((long)(n0blk + (nt + 1) * 16 + trow), (long)(OUT_DIM - 1));
            const float* src = W + grow * EMB_D + tcol;
            r0 = __builtin_nontemporal_load((const v4f*)src);
            r1 = __builtin_nontemporal_load((const v4f*)(src + 4));
        }
        __syncthreads();

        // B fragment from LDS: lane reads tile[nsub][4j+kb .. +1].
        // Padded stride 132 -> the 32 lanes hit 32 distinct bank pairs.
        const int fbase = buf + nsub * TSTRIDE + kb;

        v8f acc0 = {};
        v8f acc1 = {};
        #pragma unroll
        for (int j = 0; j < 32; ++j) {
            const v2f bj = *(const v2f*)&ldsb[fbase + 4 * j];
            acc0 = __builtin_amdgcn_wmma_f32_16x16x4_f32(
                false, a0[j], false, bj, (short)0, acc0, false, false);
            acc1 = __builtin_amdgcn_wmma_f32_16x16x4_f32(
                false, a1[j], false, bj, (short)0, acc1, false, false);
        }

        const int  col   = n0blk + nt * 16 + nsub;
        const bool valid = (col < OUT_DIM);
        if (valid) {
            const float bb = bout[col];
            #pragma unroll
            for (int r = 0; r < 8; ++r) {
                float v  = acc0[r] + bb;
                float mn = fmaxf(m0[r], v);
                s0[r] = s0[r] * __expf(m0[r] - mn) + __expf(v - mn);
                m0[r] = mn;

                v  = acc1[r] + bb;
                mn = fmaxf(m1[r], v);
                s1[r] = s1[r] * __expf(m1[r] - mn) + __expf(v - mn);
                m1[r] = mn;
            }
        }
        __syncthreads();   // tile consumed; safe to overwrite this buffer in 2 iters
    }

    // Scatter per-lane stats: row = rowbase + tile*16 + r + 8*(lane>=16).
    const int half = (lane >> 4) * 8;
    #pragma unroll
    for (int r = 0; r < 8; ++r) {
        int row = rowbase + r + half;
        smx[row * 16 + nsub] = m0[r];
        sms[row * 16 + nsub] = s0[r];
        row = rowbase + 16 + r + half;
        smx[row * 16 + nsub] = m1[r];
        sms[row * 16 + nsub] = s1[r];
    }
    __syncthreads();

    // One thread per batch row merges its 16 lane-partials (log-sum-exp merge).
    const int row = tid;
    float M = -INFINITY;
    #pragma unroll
    for (int i = 0; i < 16; ++i) M = fmaxf(M, smx[row * 16 + i]);
    float S = 0.f;
    #pragma unroll
    for (int i = 0; i < 16; ++i) S += sms[row * 16 + i] * __expf(smx[row * 16 + i] - M);

    pm[(long)blockIdx.x * BATCH + row] = M;
    ps[(long)blockIdx.x * BATCH + row] = S;
}

// ---------------------------------------------------------------------------
// Kernel 3: per-row finalize — target logit + merge NBLK partials -> row loss.
// grid = BATCH blocks (one per row), block = 256 threads.
// ---------------------------------------------------------------------------
__global__ __launch_bounds__(256)
void finalize_rows_kernel(const float* __restrict__ q,
                          const float* __restrict__ W,
                          const float* __restrict__ bout,
                          const int* __restrict__ y,
                          const float* __restrict__ pm,
                          const float* __restrict__ ps,
                          float* __restrict__ rowloss)
{
    const int row = blockIdx.x;
    const int t   = threadIdx.x;
    __shared__ float red[256];

    // Target logit: q[row] . W[y[row]] + b_out[y[row]]
    float part = 0.f;
    if (t < EMB_D) {
        const long yy = (long)y[row];
        part = q[row * EMB_D + t] * W[yy * EMB_D + t];
    }
    red[t] = part;
    __syncthreads();
    for (int off = 128; off > 0; off >>= 1) {
        if (t < off) red[t] += red[t + off];
        __syncthreads();
    }
    const float tl = red[0] + bout[y[row]];
    __syncthreads();

    // Global max over block partials.
    float lm = -INFINITY;
    for (int b = t; b < NBLK; b += 256)
        lm = fmaxf(lm, pm[(long)b * BATCH + row]);
    red[t] = lm;
    __syncthreads();
    for (int off = 128; off > 0; off >>= 1) {
        if (t < off) red[t] = fmaxf(red[t], red[t + off]);
        __syncthreads();
    }
    const float M = red[0];
    __syncthreads();

    // Global sumexp via log-sum-exp merge.
    float lssum = 0.f;
    for (int b = t; b < NBLK; b += 256)
        lssum += ps[(long)b * BATCH + row] * __expf(pm[(long)b * BATCH + row] - M);
    red[t] = lssum;
    __syncthreads();
    for (int off = 128; off > 0; off >>= 1) {
        if (t < off) red[t] += red[t + off];
        __syncthreads();
    }
    if (t == 0)
        rowloss[row] = -(tl - M - logf(red[0]));
}

// ---------------------------------------------------------------------------
// Kernel 4: mean over rows -> scalar loss (deterministic, no atomics).
// ---------------------------------------------------------------------------
__global__ __launch_bounds__(BATCH)
void mean_loss_kernel(const float* __restrict__ rowloss, float* __restrict__ out)
{
    __shared__ float red[BATCH];
    const int t = threadIdx.x;
    red[t] = rowloss[t];
    __syncthreads();
    for (int off = BATCH / 2; off > 0; off >>= 1) {
        if (t < off) red[t] += red[t + off];
        __syncthreads();
    }
    if (t == 0) out[0] = red[0] / (float)BATCH;
}

// ---------------------------------------------------------------------------
extern "C" void kernel_launch(void* const* d_in, const int* in_sizes, int n_in,
                              void* d_out, int out_size, void* d_ws, size_t ws_size,
                              hipStream_t stream)
{
    const int*   x    = (const int*)d_in[0];     // [B, T] int32
    const int*   y    = (const int*)d_in[1];     // [B] int32
    // d_in[2] = freeze, d_in[3] = slide (unused scalars)
    const float* emb  = (const float*)d_in[4];   // [IN+1, 128]
    const float* bias = (const float*)d_in[5];   // [128]
    const float* W    = (const float*)d_in[6];   // [OUT, 128]
    const float* bout = (const float*)d_in[7];   // [OUT]
    float*       out  = (float*)d_out;           // scalar loss

    float* ws      = (float*)d_ws;
    float* q       = ws;                                   // BATCH*128
    float* pm      = q  + (long)BATCH * EMB_D;             // NBLK*BATCH
    float* ps      = pm + (long)NBLK * BATCH;              // NBLK*BATCH
    float* rowloss = ps + (long)NBLK * BATCH;              // BATCH

    embed_query_kernel<<<BATCH, EMB_D, 0, stream>>>(x, emb, bias, q);
    gemm_softmax_stats_kernel<<<NBLK, 256, 0, stream>>>(q, W, bout, pm, ps);
    finalize_rows_kernel<<<BATCH, 256, 0, stream>>>(q, W, bout, y, pm, ps, rowloss);
    mean_loss_kernel<<<1, BATCH, 0, stream>>>(rowloss, out);
}